// RROIMask_85358180041456
// MI455X (gfx1250) — compile-verified
//
#include <hip/hip_runtime.h>
#include <stdint.h>

// RROIMask crop-and-resize, CDNA5 (gfx1250).
// Bandwidth-bound (512 MiB stores vs ~1 GFLOP math) -> no WMMA on purpose.
// Data movement uses the CDNA5 Tensor Data Mover (tensor_load_to_lds, one 2D
// tile DMA per block, TENSORcnt) with per-lane global_load_async_to_lds_b32
// (ASYNCcnt) as compile-time fallback.

namespace {
constexpr int kS = 2, kB = 8, kM = 16, kC = 32, kH = 128, kW = 128;
constexpr int kOut = 128;
constexpr int kSpan = 64;           // max staged window extent (rows & cols)
constexpr int kStrideMax = kSpan + 1;
// LDS: ax0/ax1/ay0/ay1 (4*128 f32), ix0/ix1/iy0/iy1 (4*128 i32), region
constexpr int kSmemBytes = (4 * 128 + 4 * 128 + kSpan * kStrideMax) * 4;
}  // namespace

#if defined(__gfx1250__) && __has_builtin(__builtin_amdgcn_tensor_load_to_lds)
#define RROI_USE_TDM 1
#else
#define RROI_USE_TDM 0
#endif

typedef unsigned int v4u __attribute__((ext_vector_type(4)));
typedef int v4i __attribute__((ext_vector_type(4)));
typedef int v8i __attribute__((ext_vector_type(8)));

__device__ __forceinline__ void async_load_f32_to_lds(const float* g,
                                                      uint32_t lds_byte) {
  asm volatile("global_load_async_to_lds_b32 %0, %1, off"
               :
               : "v"(lds_byte), "v"((uint64_t)(uintptr_t)g)
               : "memory");
}

__device__ __forceinline__ void wait_async_zero() {
#if __has_builtin(__builtin_amdgcn_s_wait_asynccnt)
  __builtin_amdgcn_s_wait_asynccnt(0);
#else
  asm volatile("s_wait_asynccnt 0" ::: "memory");
#endif
}

#if RROI_USE_TDM
// DMA a 2D f32 tile [rh x tile_w] (row stride kW elements) starting at `g`
// into LDS byte offset `ldsb` with packed LDS row stride tile_w.
// OOB columns/rows (beyond tensor_dim) are zero-filled by the TDM.
__device__ __forceinline__ void tdm_load_tile(const float* g, uint32_t ldsb,
                                              int tile_w, int rh, int rem_w,
                                              int rem_h) {
  const uint64_t ga = (uint64_t)(uintptr_t)g;
  v4u g0;
  g0.x = 1u;                 // count=1 (valid), user descriptor, no gather
  g0.y = ldsb;               // lds_addr (bytes)
  g0.z = (uint32_t)ga;       // global_addr[31:0]
  g0.w = ((uint32_t)(ga >> 32) & 0x01FFFFFFu) | (2u << 30);  // addr[56:32]|type=2
  const uint32_t td0 = (uint32_t)rem_w;   // tensor_dim0 (elements to map edge)
  const uint32_t td1 = (uint32_t)rem_h;   // tensor_dim1
  v8i g1;
  g1[0] = (int)(2u << 16);                               // data_size=4B, mask=0
  g1[1] = (int)((td0 & 0xFFFFu) << 16);                  // tensor_dim0 lo
  g1[2] = (int)((td0 >> 16) | ((td1 & 0xFFFFu) << 16));  // td0 hi | td1 lo
  g1[3] = (int)((td1 >> 16) | ((uint32_t)tile_w << 16)); // td1 hi | tile_dim0
  g1[4] = rh & 0xFFFF;                                   // tile_dim1, tile_dim2=0
  g1[5] = kW;                                            // tensor_dim0_stride
  g1[6] = 0;                                             // stride hi | dim1_stride lo
  g1[7] = 0;                                             // dim1_stride hi
  const v4i gz = {0, 0, 0, 0};                           // groups 2/3 unused (2D)
#if __has_include(<hip/amd_detail/amd_gfx1250_TDM.h>)
  const v8i gz8 = {0, 0, 0, 0, 0, 0, 0, 0};
  __builtin_amdgcn_tensor_load_to_lds(g0, g1, gz, gz, gz8, 0);  // clang-23 form
#else
  __builtin_amdgcn_tensor_load_to_lds(g0, g1, gz, gz, 0);       // ROCm 7.2 form
#endif
}
#endif  // RROI_USE_TDM

__global__ void __launch_bounds__(256) rroi_crop_kernel(
    const float* __restrict__ fm, const float* __restrict__ boxes,
    float* __restrict__ out) {
  extern __shared__ float smem[];
  float* ax0 = smem;
  float* ax1 = ax0 + 128;
  float* ay0 = ax1 + 128;
  float* ay1 = ay0 + 128;
  int* ix0 = (int*)(ay1 + 128);
  int* ix1 = ix0 + 128;
  int* iy0 = ix1 + 128;
  int* iy1 = iy0 + 128;
  float* region = (float*)(iy1 + 128);

  const int tid = threadIdx.x;
  const int bid = blockIdx.x;
  const int c = bid & (kC - 1);
  const int box = bid >> 5;  // /kC
  const int m = box & (kM - 1);
  const int b = (box >> 4) & (kB - 1);
  const int s = box >> 7;

  const float* fmc = fm + (size_t)((s * kB + b) * kC + c) * (kH * kW);
  const float* bx = boxes + (size_t)((s * kB + b) * kM + m) * 4;
  const float x1 = bx[0], y1 = bx[1], x3 = bx[2], y3 = bx[3];
  const float dx = (x3 - x1) * (1.0f / kOut);
  const float dy = (y3 - y1) * (1.0f / kOut);

  // Source window touched by the sampling grid (clamped to the map).
  int rx0 = max(0, (int)floorf(x1));
  int rxN = min(kW - 1, (int)floorf(x1 + (float)(kOut - 1) * dx) + 1);
  int ry0 = max(0, (int)floorf(y1));
  int ryN = min(kH - 1, (int)floorf(y1 + (float)(kOut - 1) * dy) + 1);
  rxN = max(rxN, rx0);
  ryN = max(ryN, ry0);
  const int rw = rxN - rx0 + 1;
  const int rh = ryN - ry0 + 1;
  const int rwp = rw | 1;  // odd LDS row stride -> spreads banks; extra col
                           // is TDM zero-fill / never read on async path
  const bool staged = (rw <= kSpan) && (rh <= kSpan);  // block-uniform

  // Separable bilinear weights + clamped taps; x on waves 0-3, y on 4-7.
  {
    const int i = tid & 127;
    const bool isY = tid >= 128;
    const float o = isY ? y1 : x1;
    const float d = isY ? dy : dx;
    const int lim = isY ? kH : kW;
    const int r0 = isY ? ry0 : rx0;
    const int rn = isY ? rh : rw;
    const float sc = o + (float)i * d;
    const float f0 = floorf(sc);
    const float w = sc - f0;
    const int p0 = (int)f0;
    const int p1 = p0 + 1;
    const float m0 = (p0 >= 0 && p0 < lim) ? 1.0f : 0.0f;  // zero border
    const float m1 = (p1 >= 0 && p1 < lim) ? 1.0f : 0.0f;
    int c0 = min(max(p0, 0), lim - 1);
    int c1 = min(max(p1, 0), lim - 1);
    if (staged) {  // window-relative taps for the LDS path
      c0 = min(max(c0 - r0, 0), rn - 1);
      c1 = min(max(c1 - r0, 0), rn - 1);
    }
    if (isY) {
      ay0[i] = (1.0f - w) * m0;
      ay1[i] = w * m1;
      iy0[i] = c0;
      iy1[i] = c1;
    } else {
      ax0[i] = (1.0f - w) * m0;
      ax1[i] = w * m1;
      ix0[i] = c0;
      ix1[i] = c1;
    }
  }

  if (staged) {
    const uint32_t base = (uint32_t)(uintptr_t)region;  // addr[31:0] = LDS off
#if RROI_USE_TDM
    if (tid == 0) {  // TDM ignores EXEC; issue the tile DMA from one wave
      tdm_load_tile(fmc + (size_t)ry0 * kW + rx0, base, rwp, rh, kW - rx0,
                    kH - ry0);
    }
#else
    const int wave = tid >> 5;
    const int lane = tid & 31;
    for (int r = wave; r < rh; r += 8) {
      const float* grow = fmc + (size_t)(ry0 + r) * kW + rx0;
      const uint32_t lrow = base + (uint32_t)(r * rwp) * 4u;
      for (int q = lane; q < rw; q += 32) {
        async_load_f32_to_lds(grow + q, lrow + (uint32_t)q * 4u);
      }
    }
#endif
  }
#if RROI_USE_TDM
  __builtin_amdgcn_s_wait_tensorcnt(0);  // wave0's DMA landed in LDS
#else
  wait_async_zero();
#endif
  __syncthreads();  // all waves: staged tile + weight tables visible

  const int u = tid & 127;
  const float a0 = ax0[u];
  const float a1 = ax1[u];
  const int cx0 = ix0[u];
  const int cx1 = ix1[u];
  float* op = out + (size_t)((s * (kB * kM) + b * kM + m) * kC + c) *
                        (size_t)(kOut * kOut);

  if (staged) {
    for (int v = (tid >> 7); v < kOut; v += 2) {
      const float* r0 = region + iy0[v] * rwp;
      const float* r1 = region + iy1[v] * rwp;
      const float fx0 = r0[cx0] * a0 + r0[cx1] * a1;
      const float fx1 = r1[cx0] * a0 + r1[cx1] * a1;
      op[v * kOut + u] = fx0 * ay0[v] + fx1 * ay1[v];
    }
  } else {  // oversized window: gather straight from L2-resident feature map
    for (int v = (tid >> 7); v < kOut; v += 2) {
      const float* r0 = fmc + (size_t)iy0[v] * kW;
      const float* r1 = fmc + (size_t)iy1[v] * kW;
      const float fx0 = r0[cx0] * a0 + r0[cx1] * a1;
      const float fx1 = r1[cx0] * a0 + r1[cx1] * a1;
      op[v * kOut + u] = fx0 * ay0[v] + fx1 * ay1[v];
    }
  }
}

extern "C" void kernel_launch(void* const* d_in, const int* in_sizes, int n_in,
                              void* d_out, int out_size, void* d_ws,
                              size_t ws_size, hipStream_t stream) {
  (void)in_sizes; (void)n_in; (void)out_size; (void)d_ws; (void)ws_size;
  const float* fm = (const float*)d_in[0];
  const float* boxes = (const float*)d_in[1];
  float* out = (float*)d_out;
  dim3 grid(kS * kB * kM * kC);  // 8192 blocks: one (s,b,m,c) crop each
  dim3 block(256);               // 8 wave32 waves
  hipLaunchKernelGGL(rroi_crop_kernel, grid, block, kSmemBytes, stream, fm,
                     boxes, out);
}